// PGNN_Layer_Base_4612794876384
// MI455X (gfx1250) — compile-verified
//
#include <hip/hip_runtime.h>
#include <math.h>

// Problem constants (from reference setup_inputs)
#define B_    8
#define NE_   4096
#define N_    1024
#define KK    50
#define F_    32
#define B2_   2048
#define ALPHA_ 0.5f

#define AS_GLOBAL __attribute__((address_space(1)))
#define AS_LDS    __attribute__((address_space(3)))

// 16-byte int vector matching the async-LDS builtins' pointee type
typedef __attribute__((__vector_size__(16))) int v4i_t;

// ---- feature availability (device pass only; host pass takes fallbacks)
#if defined(__AMDGCN__)
#define HAS_ASYNC_LDS (__has_builtin(__builtin_amdgcn_global_load_async_to_lds_b128) && \
                       __has_builtin(__builtin_amdgcn_global_store_async_from_lds_b128))
#define HAS_TDM       (__has_builtin(__builtin_amdgcn_tensor_load_to_lds) && \
                       __has_builtin(__builtin_amdgcn_tensor_store_from_lds) && \
                       __has_builtin(__builtin_amdgcn_s_wait_tensorcnt))
#else
#define HAS_ASYNC_LDS 0
#define HAS_TDM 0
#endif

typedef __attribute__((ext_vector_type(2))) float v2f;
typedef __attribute__((ext_vector_type(8))) float v8f;

// ---------- CDNA5 WMMA cross-lane sum: V_WMMA_F32_16X16X4_F32 with B = ones.
// Lane L's partial goes to A[L%16][L<16?0:2]; D[m][n] = p[m]+p[m+16]; summing the
// 8 D VGPRs + one xor-16 shuffle yields the full 32-lane sum in every lane.
// Requires EXEC all-ones (called from fully active, non-divergent code).
__device__ __forceinline__ float wave_sum32_wmma(float x) {
    v2f a; a.x = x;    a.y = 0.0f;
    v2f b; b.x = 1.0f; b.y = 1.0f;
    v8f c = {};
    c = __builtin_amdgcn_wmma_f32_16x16x4_f32(false, a, false, b, (short)0, c, false, false);
    float s = c[0] + c[1] + c[2] + c[3] + c[4] + c[5] + c[6] + c[7];
    s += __shfl_xor(s, 16, 32);
    return s;
}

__device__ __forceinline__ unsigned hash_u32(unsigned x) {
    x ^= x >> 17; x *= 0xed5ad4bbu;
    x ^= x >> 11; x *= 0xac4c1b51u;
    x ^= x >> 15; x *= 0x31848babu;
    x ^= x >> 14;
    return x;
}
__device__ __forceinline__ float hash_unif(unsigned x) {
    return (hash_u32(x) >> 8) * (1.0f / 16777216.0f);
}

#if HAS_TDM
typedef __attribute__((ext_vector_type(4))) unsigned int u32x4_t;
typedef __attribute__((ext_vector_type(4))) int          i32x4_t;
typedef __attribute__((ext_vector_type(8))) int          i32x8_t;

// D# group 0: count=1 user descriptor, lds_addr, 57-bit global addr, type=2.
__device__ __forceinline__ u32x4_t tdm_g0(const void* gaddr, unsigned lds_off) {
    unsigned long long ga = (unsigned long long)(size_t)gaddr;
    u32x4_t g0;
    g0[0] = 1u;
    g0[1] = lds_off;
    g0[2] = (unsigned)ga;
    g0[3] = (unsigned)((ga >> 32) & 0x01FFFFFFull) | (2u << 30);
    return g0;
}
// D# group 1: data_size=4B, 1 row of `nelem` f32 (tensor_dim0=tile_dim0=nelem,
// dim1=1, stride0=nelem, workgroup_mask=0 — not in a cluster).
__device__ __forceinline__ i32x8_t tdm_g1_row(int nelem) {
    i32x8_t g1;
    g1[0] = (int)(2u << 16);                              // data_size code 2 = 4B
    g1[1] = (int)(((unsigned)nelem & 0xFFFFu) << 16);     // tensor_dim0[15:0]
    g1[2] = (int)((((unsigned)nelem >> 16) & 0xFFFFu) | (1u << 16)); // dim0 hi | tensor_dim1=1 lo
    g1[3] = (int)(((unsigned)nelem & 0xFFFFu) << 16);     // tensor_dim1 hi=0 | tile_dim0
    g1[4] = 1;                                            // tile_dim1=1, tile_dim2=0
    g1[5] = nelem;                                        // tensor_dim0_stride[31:0]
    g1[6] = 0;
    g1[7] = 0;
    return g1;
}

__device__ __forceinline__ void tdm_load_row(const void* gsrc, unsigned lds_off, int nelem) {
    u32x4_t g0 = tdm_g0(gsrc, lds_off);
    i32x8_t g1 = tdm_g1_row(nelem);
    i32x4_t z4 = {};
#if __clang_major__ >= 23
    i32x8_t z8 = {};
    __builtin_amdgcn_tensor_load_to_lds(g0, g1, z4, z4, z8, 0);
#else
    __builtin_amdgcn_tensor_load_to_lds(g0, g1, z4, z4, 0);
#endif
}
__device__ __forceinline__ void tdm_store_row(void* gdst, unsigned lds_off, int nelem) {
    u32x4_t g0 = tdm_g0(gdst, lds_off);
    i32x8_t g1 = tdm_g1_row(nelem);
    i32x4_t z4 = {};
#if __clang_major__ >= 23
    i32x8_t z8 = {};
    __builtin_amdgcn_tensor_store_from_lds(g0, g1, z4, z4, z8, 0);
#else
    __builtin_amdgcn_tensor_store_from_lds(g0, g1, z4, z4, 0);
#endif
}
__device__ __forceinline__ void wait_tensor0() { __builtin_amdgcn_s_wait_tensorcnt(0); }
#endif // HAS_TDM

#if HAS_ASYNC_LDS
__device__ __forceinline__ void wait_async0() {
#if __has_builtin(__builtin_amdgcn_s_wait_asynccnt)
    __builtin_amdgcn_s_wait_asynccnt(0);
#else
    asm volatile("s_wait_asynccnt 0x0" ::: "memory");
#endif
}
#endif

// ---------------------------------------------------------------------------
// K0: zero h_w accumulator in workspace (atomics accumulate into it).
__global__ void k_zero(float* p, int n) {
    int i = blockIdx.x * blockDim.x + threadIdx.x;
    if (i < n) p[i] = 0.0f;
}

// ---------------------------------------------------------------------------
// K1: scatter-add m_w_j2i[b,e,n] into h_w[b,e_idx[e],n] via LDS accumulators.
// grid = (B_, N_/256, NE_/256), block = 64. Each thread owns a float4 column of
// a [K][256] LDS tile (51.2 KB of the 320 KB WGP LDS); all global traffic is B128.
__global__ void k_scatter_j2i(const float* __restrict__ m, const int* __restrict__ e_idx,
                              float* __restrict__ h_w) {
    __shared__ float4 acc[KK * 64];
    const int t  = threadIdx.x;          // 0..63, owns n = n0 + 4t .. 4t+3
    const int b  = blockIdx.x;
    const int n0 = blockIdx.y * 256;
    const int e0 = blockIdx.z * 256;
    const float4 z4 = make_float4(0.f, 0.f, 0.f, 0.f);
    #pragma unroll
    for (int k = 0; k < KK; ++k) acc[k * 64 + t] = z4;
    // no barrier: each thread touches only its own LDS column
    for (int e = e0; e < e0 + 256; ++e) {
        int k = e_idx[e];                // loop-uniform -> scalar load
        const float4* src = (const float4*)(m + ((size_t)(b * NE_ + e)) * N_ + n0);
        int ep = (e + 8 < NE_) ? (e + 8) : e;   // SW prefetch: no HW prefetcher on CDNA5
        __builtin_prefetch(m + ((size_t)(b * NE_ + ep)) * N_ + n0 + t * 4, 0, 0);
        float4 v = src[t];
        float4 a = acc[k * 64 + t];
        a.x += v.x; a.y += v.y; a.z += v.z; a.w += v.w;
        acc[k * 64 + t] = a;
    }
    for (int k = 0; k < KK; ++k) {
        float* dst = h_w + ((size_t)(b * KK + k)) * N_ + n0 + t * 4;
        float4 a = acc[k * 64 + t];
        atomicAdd(dst + 0, a.x); atomicAdd(dst + 1, a.y);
        atomicAdd(dst + 2, a.z); atomicAdd(dst + 3, a.w);
    }
}

// K2: add m_w_o2i[b,o,n] into h_w[b,o_idx[o],n]. grid=(KK-1, B_), block=256 (float4/thread).
__global__ void k_scatter_o2i(const float* __restrict__ m, const int* __restrict__ o_idx,
                              float* __restrict__ h_w) {
    const int t = threadIdx.x;
    const int o = blockIdx.x;
    const int b = blockIdx.y;
    int k = o_idx[o];
    float4 v = ((const float4*)(m + ((size_t)(b * (KK - 1) + o)) * N_))[t];
    float* dst = h_w + ((size_t)(b * KK + k)) * N_ + t * 4;
    atomicAdd(dst + 0, v.x); atomicAdd(dst + 1, v.y);
    atomicAdd(dst + 2, v.z); atomicAdd(dst + 3, v.w);
}

// ---------------------------------------------------------------------------
// K3: per (b,k) row: lse over n=1024, write h_w_n = h_w - lse. WMMA sum path.
// grid = B_*KK rows, block = 256 (8 waves); one B128 load/store per thread.
__global__ void k_lse_rows(const float* __restrict__ h_w, float* __restrict__ out0) {
    const int row  = blockIdx.x;
    const int tid  = threadIdx.x;
    const int lane = tid & 31;
    const int wid  = tid >> 5;

    __shared__ float smax[8];
    __shared__ float ssum[8];

    float4 v = ((const float4*)(h_w + (size_t)row * N_))[tid];

    // max
    float m = fmaxf(fmaxf(v.x, v.y), fmaxf(v.z, v.w));
    #pragma unroll
    for (int off = 16; off > 0; off >>= 1) m = fmaxf(m, __shfl_xor(m, off, 32));
    if (lane == 0) smax[wid] = m;
    __syncthreads();
    float M = smax[0];
    #pragma unroll
    for (int j = 1; j < 8; ++j) M = fmaxf(M, smax[j]);

    // sum of exp, 32-lane reduction through V_WMMA_F32_16X16X4_F32
    float p = expf(v.x - M) + expf(v.y - M) + expf(v.z - M) + expf(v.w - M);
    float ws = wave_sum32_wmma(p);
    if (lane == 0) ssum[wid] = ws;
    __syncthreads();
    float S = 0.0f;
    #pragma unroll
    for (int j = 0; j < 8; ++j) S += ssum[j];
    float lse = M + logf(S);

    float4 r = make_float4(v.x - lse, v.y - lse, v.z - lse, v.w - lse);
    ((float4*)(out0 + (size_t)row * N_))[tid] = r;
}

// ---------------------------------------------------------------------------
// K4: feature gather: out1[b,e,:,:] = features[b, edge_idx[b,e], :, :]
// Each wave copies one 6.4 KB row. TDM path: program a D# and let the Tensor
// Data Mover DMA Global->LDS->Global (TENSORcnt). Fallback: direct B128 copy.
// grid = B_*NE_/8 blocks, block = 256 (8 waves).
__global__ void k_gather_feat(const float* __restrict__ feat, const int* __restrict__ edge_idx,
                              float* __restrict__ out1) {
#if HAS_TDM
    __shared__ __align__(16) float tile[8 * KK * F_];     // 8 waves x 1600 f32 = 51.2 KB
    const int wid = threadIdx.x >> 5;
    const int be  = blockIdx.x * 8 + wid;                 // uniform per wave
    const int b   = be >> 12;
    const int e   = be & (NE_ - 1);
    const int node = edge_idx[b * NE_ + e];
    const float* gsrc = feat + ((size_t)b * N_ + node) * (KK * F_);
    float*       gdst = out1 + ((size_t)b * NE_ + e) * (KK * F_);
    unsigned lds_off = (unsigned)(size_t)(AS_LDS float*)&tile[wid * (KK * F_)];
    tdm_load_row(gsrc, lds_off, KK * F_);
    wait_tensor0();                                        // LDS tile resident
    tdm_store_row(gdst, lds_off, KK * F_);
    wait_tensor0();
#else
    const int tid = threadIdx.x;
    for (int r = 0; r < 8; ++r) {
        const int be  = blockIdx.x * 8 + r;
        const int b   = be >> 12;
        const int e   = be & (NE_ - 1);
        const int node = edge_idx[b * NE_ + e];
        const float4* src = (const float4*)(feat + ((size_t)b * N_ + node) * (KK * F_));
        float4*       dst = (float4*)(out1 + ((size_t)b * NE_ + e) * (KK * F_));
        for (int i = tid; i < (KK * F_) / 4; i += 256) dst[i] = src[i];
    }
#endif
}

// ---------------------------------------------------------------------------
// K5: soft resampling: per row i of B2, K draws from CDF of
// alpha*exp(prob)+(1-alpha)/K, emit flat indices + normalized log-weights.
// grid = B2_, block = 64.
__global__ void k_resample(const float* __restrict__ prob, int* __restrict__ flat_ws,
                           float* __restrict__ out3) {
    __shared__ float cdf[KK];
    __shared__ float red[64];
    __shared__ float tot_s;
    const int i   = blockIdx.x;
    const int tid = threadIdx.x;

    if (tid < KK)
        cdf[tid] = ALPHA_ * expf(prob[i * KK + tid]) + (1.0f - ALPHA_) / (float)KK;
    __syncthreads();
    if (tid == 0) {
        float run = 0.0f;
        for (int k = 0; k < KK; ++k) { run += cdf[k]; cdf[k] = run; }
        tot_s = run;
    }
    __syncthreads();

    float lp = -3.402823466e38f;
    if (tid < KK) {
        float u = hash_unif((unsigned)(i * KK + tid) * 2654435761u + 17u) * tot_s;
        int idx = 0;
        while (idx < KK - 1 && cdf[idx] < u) ++idx;
        int flat = i + idx * B2_;                 // particle-major flat index
        flat_ws[i * KK + tid] = flat;
        float pv = expf(prob[flat]) + 1e-8f;
        pv = pv / (ALPHA_ * pv + (1.0f - ALPHA_) / (float)KK);
        lp = logf(pv);
    }

    red[tid] = lp;
    __syncthreads();
    #pragma unroll
    for (int off = 32; off > 0; off >>= 1) {
        if (tid < off) red[tid] = fmaxf(red[tid], red[tid + off]);
        __syncthreads();
    }
    float M = red[0];
    __syncthreads();
    red[tid] = (tid < KK) ? expf(lp - M) : 0.0f;
    __syncthreads();
    #pragma unroll
    for (int off = 32; off > 0; off >>= 1) {
        if (tid < off) red[tid] += red[tid + off];
        __syncthreads();
    }
    float lse = M + logf(red[0]);
    if (tid < KK) out3[i * KK + tid] = lp - lse;
}

// ---------------------------------------------------------------------------
// K6: particle gather: 512 B per row, one wave each. ASYNCcnt path: each lane
// async-copies its B128 Global->LDS then LDS->Global with no VGPR round-trip.
// grid = (K*B2)/8 blocks, block = 256 (8 waves).
__global__ void k_gather_particles(const float* __restrict__ particles,
                                   const int* __restrict__ flat_ws,
                                   float* __restrict__ out2) {
    const int r    = blockIdx.x * 8 + (threadIdx.x >> 5);
    const int lane = threadIdx.x & 31;
    const int src  = flat_ws[r];
#if HAS_ASYNC_LDS
    __shared__ __align__(16) float tile[8 * 128];          // 4 KB
    AS_LDS float* lf = (AS_LDS float*)&tile[(threadIdx.x >> 5) * 128 + lane * 4];
    AS_LDS v4i_t* lv = (AS_LDS v4i_t*)lf;
    const float* gs = particles + (size_t)src * 128 + lane * 4;
    float*       gd = out2 + (size_t)r * 128 + lane * 4;
    __builtin_amdgcn_global_load_async_to_lds_b128((v4i_t*)gs, lv, 0, 0);
    wait_async0();                                         // own wave's loads done
    __builtin_amdgcn_global_store_async_from_lds_b128((v4i_t*)gd, lv, 0, 0);
    wait_async0();
#else
    ((float4*)out2)[(size_t)r * 32 + lane] =
        ((const float4*)particles)[(size_t)src * 32 + lane];
#endif
}

// ---------------------------------------------------------------------------
// K7: reparameterize: out4 = mu + eps * softplus(var), eps via hashed Box-Muller.
// B128 loads/stores, 4 elements per thread.
__global__ void k_reparam(const float4* __restrict__ mu4, const float4* __restrict__ var4,
                          float4* __restrict__ out4, int n4) {
    int i = blockIdx.x * blockDim.x + threadIdx.x;
    if (i >= n4) return;
    float4 mv = mu4[i];
    float4 vv = var4[i];
    float vin[4]  = { vv.x, vv.y, vv.z, vv.w };
    float min_[4] = { mv.x, mv.y, mv.z, mv.w };
    float rr[4];
    #pragma unroll
    for (int j = 0; j < 4; ++j) {
        unsigned ei = (unsigned)i * 4u + (unsigned)j;
        float v  = vin[j];
        float sp = (v > 20.0f) ? v : log1pf(expf(v));
        float u1 = hash_unif(ei * 2u + 12345u) + 1e-7f;
        float u2 = hash_unif(ei * 2u + 1u + 98765u);
        float eps = sqrtf(-2.0f * logf(u1)) * cosf(6.283185307f * u2);
        rr[j] = min_[j] + eps * sp;
    }
    out4[i] = make_float4(rr[0], rr[1], rr[2], rr[3]);
}

// ---------------------------------------------------------------------------
extern "C" void kernel_launch(void* const* d_in, const int* in_sizes, int n_in,
                              void* d_out, int out_size, void* d_ws, size_t ws_size,
                              hipStream_t stream) {
    const float* m_w_j2i  = (const float*)d_in[0];
    const float* m_w_o2i  = (const float*)d_in[1];
    const float* features = (const float*)d_in[2];
    const float* particles= (const float*)d_in[3];
    const float* prob     = (const float*)d_in[4];
    const float* mu       = (const float*)d_in[5];
    const float* var      = (const float*)d_in[6];
    const int*   e_idx    = (const int*)d_in[7];
    const int*   o_idx    = (const int*)d_in[8];
    const int*   edge_idx = (const int*)d_in[9];

    // Output layout (flat, return order)
    float* out0 = (float*)d_out;                       // h_w_n      [8,50,1024,1]
    float* out1 = out0 + (size_t)B_ * KK * N_;         // gathered   [8,4096,50,32]
    float* out2 = out1 + (size_t)B_ * NE_ * KK * F_;   // particles  [102400,128]
    float* out3 = out2 + (size_t)KK * B2_ * 128;       // prob_new   [102400,1]
    float* out4 = out3 + (size_t)KK * B2_;             // reparam    [102400,128]

    // Workspace: h_w accumulator + resample flat indices
    float* h_w   = (float*)d_ws;                       // 409,600 f32
    int* flat_ws = (int*)((char*)d_ws + (size_t)B_ * KK * N_ * sizeof(float));

    const int hw_elems = B_ * KK * N_;                 // 409,600
    k_zero<<<(hw_elems + 255) / 256, 256, 0, stream>>>(h_w, hw_elems);

    k_scatter_j2i<<<dim3(B_, N_ / 256, NE_ / 256), 64, 0, stream>>>(m_w_j2i, e_idx, h_w);
    k_scatter_o2i<<<dim3(KK - 1, B_), 256, 0, stream>>>(m_w_o2i, o_idx, h_w);
    k_lse_rows<<<B_ * KK, 256, 0, stream>>>(h_w, out0);

    k_gather_feat<<<(B_ * NE_) / 8, 256, 0, stream>>>(features, edge_idx, out1);

    k_resample<<<B2_, 64, 0, stream>>>(prob, flat_ws, out3);
    k_gather_particles<<<(KK * B2_) / 8, 256, 0, stream>>>(particles, flat_ws, out2);

    const int rep_n4 = (KK * B2_ * 128) / 4;           // 3,276,800 float4
    k_reparam<<<(rep_n4 + 255) / 256, 256, 0, stream>>>(
        (const float4*)mu, (const float4*)var, (float4*)out4, rep_n4);
}